// ReformerDec_89653147337029
// MI455X (gfx1250) — compile-verified
//
#include <hip/hip_runtime.h>
#include <math.h>

typedef __attribute__((ext_vector_type(2))) float v2f;
typedef __attribute__((ext_vector_type(8))) float v8f;

__device__ __forceinline__ v8f wmma_f32(v2f a, v2f b, v8f c) {
  // V_WMMA_F32_16X16X4_F32 : D = A(16x4 f32) x B(4x16 f32) + C(16x16 f32)
  return __builtin_amdgcn_wmma_f32_16x16x4_f32(
      /*neg_a=*/false, a, /*neg_b=*/false, b,
      /*c_mod=*/(short)0, c, /*reuse_a=*/false, /*reuse_b=*/false);
}

__device__ __forceinline__ float gelu_f(float x) {
  float x3 = x * x * x;
  return 0.5f * x * (1.0f + tanhf(0.7978845608028654f * (x + 0.044715f * x3)));
}

// ---------------------------------------------------------------------------
// LayerNorm over last dim 512. One block per row, 256 threads, 2 elems/thread.
// ---------------------------------------------------------------------------
__global__ __launch_bounds__(256) void ln_kernel(
    const float* __restrict__ x, const float* __restrict__ g,
    const float* __restrict__ b, float* __restrict__ y) {
  __shared__ float red[256];
  int row = blockIdx.x;
  int tid = threadIdx.x;
  const float* xr = x + (size_t)row * 512;
  float v0 = xr[tid], v1 = xr[tid + 256];
  red[tid] = v0 + v1;
  __syncthreads();
  for (int s = 128; s > 0; s >>= 1) {
    if (tid < s) red[tid] += red[tid + s];
    __syncthreads();
  }
  float mean = red[0] * (1.0f / 512.0f);
  __syncthreads();
  float d0 = v0 - mean, d1 = v1 - mean;
  red[tid] = d0 * d0 + d1 * d1;
  __syncthreads();
  for (int s = 128; s > 0; s >>= 1) {
    if (tid < s) red[tid] += red[tid + s];
    __syncthreads();
  }
  float r = rsqrtf(red[0] * (1.0f / 512.0f) + 1e-5f);
  float* yr = y + (size_t)row * 512;
  yr[tid]       = d0 * r * g[tid]       + b[tid];
  yr[tid + 256] = d1 * r * g[tid + 256] + b[tid + 256];
}

// ---------------------------------------------------------------------------
// Generic f32 WMMA GEMM: C = act(A[M,K] @ W[K,N] + bias) + res
// 64x64 block tile, BK=16 LDS staging, 8 waves, each wave: two 16x16 C tiles.
// ---------------------------------------------------------------------------
__global__ __launch_bounds__(256) void gemm_k(
    const float* __restrict__ A, const float* __restrict__ W,
    const float* __restrict__ bias, const float* __restrict__ res,
    float* __restrict__ C,
    int M, int N, int K, int ldA, int ldW, int ldC, int act) {
  __shared__ float As[64][17];
  __shared__ float Bs[16][68];
  int tid = threadIdx.x;
  int lane = tid & 31, w = tid >> 5;
  int wm = w & 3, wn = w >> 2;  // 4 x 2 wave grid
  int rowBase = blockIdx.y * 64;
  int colBase = blockIdx.x * 64;
  int ml = lane & 15;
  int kb2 = (lane >> 4) << 1;
  int half = lane >> 4;
  v8f acc0 = {0, 0, 0, 0, 0, 0, 0, 0};
  v8f acc1 = {0, 0, 0, 0, 0, 0, 0, 0};
  for (int k0 = 0; k0 < K; k0 += 16) {
#pragma unroll
    for (int i = 0; i < 4; i++) {
      int p = tid + 256 * i;
      int m = p >> 4, kk = p & 15;
      As[m][kk] = A[(size_t)(rowBase + m) * ldA + k0 + kk];
      int kb = p >> 6, nn = p & 63;
      Bs[kb][nn] = W[(size_t)(k0 + kb) * ldW + colBase + nn];
    }
    __syncthreads();
#pragma unroll
    for (int kk = 0; kk < 16; kk += 4) {
      v2f a, b0, b1;
      a.x = As[wm * 16 + ml][kk + kb2];
      a.y = As[wm * 16 + ml][kk + kb2 + 1];
      b0.x = Bs[kk + kb2][wn * 16 + ml];
      b0.y = Bs[kk + kb2 + 1][wn * 16 + ml];
      b1.x = Bs[kk + kb2][wn * 16 + 32 + ml];
      b1.y = Bs[kk + kb2 + 1][wn * 16 + 32 + ml];
      acc0 = wmma_f32(a, b0, acc0);
      acc1 = wmma_f32(a, b1, acc1);
    }
    __syncthreads();
  }
#pragma unroll
  for (int t = 0; t < 2; t++) {
    v8f acc = t ? acc1 : acc0;
    int gc = colBase + wn * 16 + t * 32 + ml;
#pragma unroll
    for (int vi = 0; vi < 8; vi++) {
      int gr = rowBase + wm * 16 + half * 8 + vi;
      float val = acc[vi];
      if (bias) val += bias[gc];
      if (act == 1) val = gelu_f(val);
      size_t off = (size_t)gr * ldC + gc;
      if (res) val += res[off];
      C[off] = val;
    }
  }
}

// ---------------------------------------------------------------------------
// LSH bucket assignment + stable counting sort, per (bh, hash).
// grid = 16*4 blocks, 256 threads.  qk layout: proj[b][t][h8*64+d], b=bh>>3.
// ---------------------------------------------------------------------------
__global__ __launch_bounds__(256) void lsh_bucket_sort_k(
    const float* __restrict__ qk, const float* __restrict__ rotw,
    int* __restrict__ sidx, int* __restrict__ sinv) {
  __shared__ float s_rw[64 * 32];
  __shared__ unsigned char s_bucket[4096];
  __shared__ int s_cnt[64];
  __shared__ int s_off[64];
  int bh = blockIdx.x >> 2;
  int hash = blockIdx.x & 3;
  int tid = threadIdx.x;
  for (int p = tid; p < 2048; p += 256) s_rw[p] = rotw[hash * 2048 + p];
  if (tid < 64) s_cnt[tid] = 0;
  __syncthreads();
  int b = bh >> 3, h8 = bh & 7;
  for (int t = tid; t < 4096; t += 256) {
    const float* q = qk + ((size_t)b * 4096 + t) * 512 + h8 * 64;
    float accv[32];
#pragma unroll
    for (int r = 0; r < 32; r++) accv[r] = 0.f;
    for (int d = 0; d < 64; d++) {
      float qd = q[d];
#pragma unroll
      for (int r = 0; r < 32; r++) accv[r] += qd * s_rw[d * 32 + r];
    }
    // argmax over [rot, -rot], first-max tie rule like jnp.argmax
    float best = accv[0];
    int bi = 0;
    for (int j = 1; j < 32; j++)
      if (accv[j] > best) { best = accv[j]; bi = j; }
    for (int j = 0; j < 32; j++) {
      float vneg = -accv[j];
      if (vneg > best) { best = vneg; bi = 32 + j; }
    }
    s_bucket[t] = (unsigned char)bi;
    atomicAdd(&s_cnt[bi], 1);
  }
  __syncthreads();
  if (tid == 0) {
    int run = 0;
    for (int j = 0; j < 64; j++) { s_off[j] = run; run += s_cnt[j]; }
  }
  __syncthreads();
  if (tid < 64) {  // stable scatter: one thread per bucket scans positions
    int o = s_off[tid];
    size_t seg = (size_t)blockIdx.x * 4096;
    for (int t = 0; t < 4096; t++) {
      if (s_bucket[t] == (unsigned char)tid) {
        sidx[seg + o] = t;
        sinv[seg + t] = o;
        o++;
      }
    }
  }
}

// ---------------------------------------------------------------------------
// LSH chunk attention: one block per (bh, global chunk c in [0,256)).
// bq (64x64), keys = normalized [cur;prev] (128x64), causal/self masks,
// softmax with lse, bo = p@bv.  WMMA f32 for both GEMMs.
// ---------------------------------------------------------------------------
__global__ __launch_bounds__(256) void lsh_attn_k(
    const float* __restrict__ qk, const float* __restrict__ vproj,
    const int* __restrict__ sidx, float* __restrict__ so,
    float* __restrict__ sl) {
  __shared__ float s_q[64][68];
  __shared__ float s_dots[64][130];
  __shared__ float s_invn[128];
  __shared__ int s_qt[64];
  __shared__ int s_kt[128];
  __shared__ int s_kb[128];
  int bh = blockIdx.x >> 8;
  int c = blockIdx.x & 255;
  int hash = c >> 6;
  int tid = threadIdx.x;
  int b = bh >> 3, h8 = bh & 7;
  if (tid < 64) {
    int j = (c & 63) * 64 + tid;
    s_qt[tid] = sidx[((size_t)bh * 4 + hash) * 4096 + j];
  }
  if (tid < 128) {
    int cc = (tid < 64) ? c : ((c + 255) & 255);  // look_one_back (wraps)
    int m = tid & 63;
    int hh = cc >> 6;
    int j = (cc & 63) * 64 + m;
    int pos = sidx[((size_t)bh * 4 + hh) * 4096 + j];
    s_kt[tid] = pos;
    int base = (int)(((size_t)b * 4096 + pos) * 512) + h8 * 64;
    s_kb[tid] = base;
    float ss = 0.f;
    for (int d = 0; d < 64; d++) { float v = qk[base + d]; ss += v * v; }
    s_invn[tid] = 1.0f / fmaxf(sqrtf(ss), 1e-12f);
  }
  __syncthreads();
  for (int p = tid; p < 64 * 64; p += 256) {
    int m = p >> 6, d = p & 63;
    s_q[m][d] = qk[((size_t)b * 4096 + s_qt[m]) * 512 + h8 * 64 + d];
  }
  __syncthreads();
  int lane = tid & 31, w = tid >> 5;
  int ml = lane & 15, kb2 = (lane >> 4) << 1, half = lane >> 4;
  // dots = bq @ bk^T * 1/8 : 64x128, K=64 -> 32 tiles, 4 per wave
  for (int ti = w; ti < 32; ti += 8) {
    int tm = ti & 3, tn = ti >> 2;
    int n = tn * 16 + ml;  // key index j
    int kbase = s_kb[n];
    float invn = s_invn[n];
    v8f acc = {0, 0, 0, 0, 0, 0, 0, 0};
#pragma unroll
    for (int kk = 0; kk < 64; kk += 4) {
      v2f a, bb;
      a.x = s_q[tm * 16 + ml][kk + kb2];
      a.y = s_q[tm * 16 + ml][kk + kb2 + 1];
      bb.x = qk[kbase + kk + kb2] * invn;
      bb.y = qk[kbase + kk + kb2 + 1] * invn;
      acc = wmma_f32(a, bb, acc);
    }
#pragma unroll
    for (int vi = 0; vi < 8; vi++)
      s_dots[tm * 16 + half * 8 + vi][n] = acc[vi] * 0.125f;
  }
  __syncthreads();
  // masked softmax per query row (64 threads)
  if (tid < 64) {
    int qpos = s_qt[tid];
    float mx = -3.4e38f;
    for (int j = 0; j < 128; j++) {
      float dd = s_dots[tid][j];
      int kpos = s_kt[j];
      if (qpos < kpos) dd = -1e9f;         // causal mask
      else if (qpos == kpos) dd = -5e4f;   // no self-attention
      s_dots[tid][j] = dd;
      mx = fmaxf(mx, dd);
    }
    float sum = 0.f;
    for (int j = 0; j < 128; j++) sum += expf(s_dots[tid][j] - mx);
    float lse = mx + logf(sum);
    for (int j = 0; j < 128; j++) s_dots[tid][j] = expf(s_dots[tid][j] - lse);
    sl[(size_t)bh * 16384 + c * 64 + tid] = lse;
  }
  __syncthreads();
  // bo = p @ bv : 64x64, K=128 -> 16 tiles, 2 per wave
  for (int ti = w; ti < 16; ti += 8) {
    int tm = ti & 3, tn = ti >> 2;
    int n = tn * 16 + ml;  // d index
    v8f acc = {0, 0, 0, 0, 0, 0, 0, 0};
#pragma unroll 4
    for (int kk = 0; kk < 128; kk += 4) {
      v2f a, bb;
      a.x = s_dots[tm * 16 + ml][kk + kb2];
      a.y = s_dots[tm * 16 + ml][kk + kb2 + 1];
      bb.x = vproj[s_kb[kk + kb2] + n];
      bb.y = vproj[s_kb[kk + kb2 + 1] + n];
      acc = wmma_f32(a, bb, acc);
    }
#pragma unroll
    for (int vi = 0; vi < 8; vi++) {
      int m = tm * 16 + half * 8 + vi;
      so[((size_t)bh * 16384 + c * 64 + m) * 64 + n] = acc[vi];
    }
  }
}

// ---------------------------------------------------------------------------
// Unsort + combine 4 hash rounds.  grid = 16*4096 blocks x 64 threads.
// Writes merged-head layout out[b][t][h8*64+d].
// ---------------------------------------------------------------------------
__global__ __launch_bounds__(64) void lsh_combine_k(
    const float* __restrict__ so, const float* __restrict__ sl,
    const int* __restrict__ sinv, float* __restrict__ out) {
  int bh = blockIdx.x >> 12;
  int t = blockIdx.x & 4095;
  int d = threadIdx.x;
  int b = bh >> 3, h8 = bh & 7;
  float lg[4];
  int fl[4];
  float mx = -3.4e38f;
#pragma unroll
  for (int h = 0; h < 4; h++) {
    int j = sinv[((size_t)bh * 4 + h) * 4096 + t];
    int flat = h * 4096 + j;
    fl[h] = flat;
    lg[h] = sl[(size_t)bh * 16384 + flat];
    mx = fmaxf(mx, lg[h]);
  }
  float sum = 0.f;
#pragma unroll
  for (int h = 0; h < 4; h++) sum += expf(lg[h] - mx);
  float lse = mx + logf(sum);
  float o = 0.f;
#pragma unroll
  for (int h = 0; h < 4; h++) {
    float wgt = expf(lg[h] - lse);
    o += wgt * so[((size_t)bh * 16384 + fl[h]) * 64 + d];
  }
  out[((size_t)b * 4096 + t) * 512 + h8 * 64 + d] = o;
}

// ---------------------------------------------------------------------------
// Flash-style cross attention.  grid = (b*8+h)*64 + qtile blocks (1024).
// Streams 1024 keys in chunks of 128 with online softmax; WMMA accumulators.
// ---------------------------------------------------------------------------
__global__ __launch_bounds__(256) void mha_flash_k(
    const float* __restrict__ q, const float* __restrict__ k,
    const float* __restrict__ v, float* __restrict__ out) {
  __shared__ float s_q[64][68];
  __shared__ float s_dots[64][130];
  __shared__ float s_max[64], s_sum[64], s_scale[64];
  int qt = blockIdx.x & 63;
  int bh = blockIdx.x >> 6;
  int b = bh >> 3, h8 = bh & 7;
  int tid = threadIdx.x;
  int t0 = qt * 64;
  for (int p = tid; p < 64 * 64; p += 256) {
    int m = p >> 6, d = p & 63;
    s_q[m][d] = q[((size_t)b * 4096 + t0 + m) * 512 + h8 * 64 + d];
  }
  if (tid < 64) { s_max[tid] = -3.4e38f; s_sum[tid] = 0.f; }
  __syncthreads();
  int lane = tid & 31, w = tid >> 5;
  int ml = lane & 15, kb2 = (lane >> 4) << 1, half = lane >> 4;
  v8f accA = {0, 0, 0, 0, 0, 0, 0, 0};
  v8f accB = {0, 0, 0, 0, 0, 0, 0, 0};
  for (int kc = 0; kc < 1024; kc += 128) {
    // scores 64x128 over K=64
    for (int ti = w; ti < 32; ti += 8) {
      int tm = ti & 3, tn = ti >> 2;
      int n = tn * 16 + ml;
      size_t kb = ((size_t)b * 1024 + kc + n) * 512 + h8 * 64;
      v8f acc = {0, 0, 0, 0, 0, 0, 0, 0};
#pragma unroll
      for (int kk = 0; kk < 64; kk += 4) {
        v2f a, bb;
        a.x = s_q[tm * 16 + ml][kk + kb2];
        a.y = s_q[tm * 16 + ml][kk + kb2 + 1];
        bb.x = k[kb + kk + kb2];
        bb.y = k[kb + kk + kb2 + 1];
        acc = wmma_f32(a, bb, acc);
      }
#pragma unroll
      for (int vi = 0; vi < 8; vi++)
        s_dots[tm * 16 + half * 8 + vi][n] = acc[vi] * 0.125f;
    }
    __syncthreads();
    if (tid < 64) {  // online softmax per row
      float mx = s_max[tid];
      float cm = -3.4e38f;
      for (int j = 0; j < 128; j++) cm = fmaxf(cm, s_dots[tid][j]);
      float nm = fmaxf(mx, cm);
      float factor = expf(mx - nm);
      float sum = 0.f;
      for (int j = 0; j < 128; j++) {
        float p = expf(s_dots[tid][j] - nm);
        s_dots[tid][j] = p;
        sum += p;
      }
      s_sum[tid] = s_sum[tid] * factor + sum;
      s_max[tid] = nm;
      s_scale[tid] = factor;
    }
    __syncthreads();
    // o = o*factor + p @ v_chunk (K=128), 2 tiles per wave
#pragma unroll
    for (int t2 = 0; t2 < 2; t2++) {
      int ti = w + t2 * 8;
      int tm = ti & 3, tn = ti >> 2;
      v8f acc = t2 ? accB : accA;
#pragma unroll
      for (int vi = 0; vi < 8; vi++) acc[vi] *= s_scale[tm * 16 + half * 8 + vi];
      int n = tn * 16 + ml;
#pragma unroll 4
      for (int kk = 0; kk < 128; kk += 4) {
        v2f a, bb;
        a.x = s_dots[tm * 16 + ml][kk + kb2];
        a.y = s_dots[tm * 16 + ml][kk + kb2 + 1];
        size_t vb0 = ((size_t)b * 1024 + kc + kk + kb2) * 512 + h8 * 64;
        bb.x = v[vb0 + n];
        bb.y = v[vb0 + 512 + n];
        acc = wmma_f32(a, bb, acc);
      }
      if (t2) accB = acc; else accA = acc;
    }
    __syncthreads();
  }
#pragma unroll
  for (int t2 = 0; t2 < 2; t2++) {
    int ti = w + t2 * 8;
    int tm = ti & 3, tn = ti >> 2;
    v8f acc = t2 ? accB : accA;
    int n = tn * 16 + ml;
#pragma unroll
    for (int vi = 0; vi < 8; vi++) {
      int m = tm * 16 + half * 8 + vi;
      out[((size_t)b * 4096 + t0 + m) * 512 + h8 * 64 + n] = acc[vi] / s_sum[m];
    }
  }
}

// ---------------------------------------------------------------------------
__global__ __launch_bounds__(256) void init_k(const float* __restrict__ x,
                                              float* __restrict__ x1,
                                              float* __restrict__ x2, size_t n) {
  size_t i = (size_t)blockIdx.x * 256 + threadIdx.x;
  if (i < n) { float v = x[i]; x1[i] = v; x2[i] = v; }
}
__global__ __launch_bounds__(256) void add_k(const float* __restrict__ a,
                                             const float* __restrict__ b,
                                             float* __restrict__ o, size_t n) {
  size_t i = (size_t)blockIdx.x * 256 + threadIdx.x;
  if (i < n) o[i] = a[i] + b[i];
}

// ---------------------------------------------------------------------------
extern "C" void kernel_launch(void* const* d_in, const int* in_sizes, int n_in,
                              void* d_out, int out_size, void* d_ws,
                              size_t ws_size, hipStream_t stream) {
  const float* x        = (const float*)d_in[0];
  const float* keys     = (const float*)d_in[1];
  const float* ln_sa_g  = (const float*)d_in[2];
  const float* ln_sa_b  = (const float*)d_in[3];
  const float* ln_ca_g  = (const float*)d_in[4];
  const float* ln_ca_b  = (const float*)d_in[5];
  const float* ln_ff_g  = (const float*)d_in[6];
  const float* ln_ff_b  = (const float*)d_in[7];
  const float* sa_wqk   = (const float*)d_in[8];
  const float* sa_wv    = (const float*)d_in[9];
  const float* sa_wo    = (const float*)d_in[10];
  const float* sa_bo    = (const float*)d_in[11];
  const float* rot      = (const float*)d_in[12];
  const float* ca_in_w  = (const float*)d_in[13];
  const float* ca_in_b  = (const float*)d_in[14];
  const float* ca_out_w = (const float*)d_in[15];
  const float* ca_out_b = (const float*)d_in[16];
  const float* ff_w1    = (const float*)d_in[17];
  const float* ff_b1    = (const float*)d_in[18];
  const float* ff_w2    = (const float*)d_in[19];
  const float* ff_b2    = (const float*)d_in[20];

  float* ws = (float*)d_ws;
  const size_t SZ_X = (size_t)8192 * 512;        // 4,194,304 floats
  float* X1 = ws;
  float* X2 = X1 + SZ_X;
  float* LN = X2 + SZ_X;
  float* PA = LN + SZ_X;                          // qk proj / q proj
  float* PB = PA + SZ_X;                          // v proj / mha pre-out
  float* KB = PB + SZ_X;                          // k proj (2048x512)
  float* VB = KB + (size_t)2048 * 512;            // v proj (2048x512)
  float* SL = VB + (size_t)2048 * 512;            // lse [16][16384]
  int* SIDX = (int*)(SL + (size_t)16 * 16384);
  int* SINV = SIDX + (size_t)16 * 16384;
  float* BIG = (float*)(SINV + (size_t)16 * 16384);  // so / ffn hidden (64MB)

  init_k<<<(int)((SZ_X + 255) / 256), 256, 0, stream>>>(x, X1, X2, SZ_X);
  float* cur1 = X1;
  float* cur2 = X2;
  for (int i = 0; i < 2; i++) {
    // ---- LSH self-attention sublayer ----
    ln_kernel<<<8192, 256, 0, stream>>>(cur2, ln_sa_g + i * 512, ln_sa_b + i * 512, LN);
    gemm_k<<<dim3(8, 128), 256, 0, stream>>>(LN, sa_wqk + (size_t)i * 512 * 512,
        nullptr, nullptr, PA, 8192, 512, 512, 512, 512, 512, 0);
    gemm_k<<<dim3(8, 128), 256, 0, stream>>>(LN, sa_wv + (size_t)i * 512 * 512,
        nullptr, nullptr, PB, 8192, 512, 512, 512, 512, 512, 0);
    lsh_bucket_sort_k<<<64, 256, 0, stream>>>(PA, rot + (size_t)i * 4 * 64 * 32, SIDX, SINV);
    lsh_attn_k<<<16 * 256, 256, 0, stream>>>(PA, PB, SIDX, BIG, SL);
    lsh_combine_k<<<16 * 4096, 64, 0, stream>>>(BIG, SL, SINV, LN);
    gemm_k<<<dim3(8, 128), 256, 0, stream>>>(LN, sa_wo + (size_t)i * 512 * 512,
        sa_bo + i * 512, cur1, cur1, 8192, 512, 512, 512, 512, 512, 0);
    { float* t = cur1; cur1 = cur2; cur2 = t; }
    // ---- cross-attention sublayer ----
    ln_kernel<<<8192, 256, 0, stream>>>(cur2, ln_ca_g + i * 512, ln_ca_b + i * 512, LN);
    const float* inW = ca_in_w + (size_t)i * 512 * 1536;
    const float* inB = ca_in_b + (size_t)i * 1536;
    gemm_k<<<dim3(8, 128), 256, 0, stream>>>(LN, inW, inB, nullptr, PA,
        8192, 512, 512, 512, 1536, 512, 0);
    gemm_k<<<dim3(8, 32), 256, 0, stream>>>(keys, inW + 512, inB + 512, nullptr, KB,
        2048, 512, 512, 512, 1536, 512, 0);
    gemm_k<<<dim3(8, 32), 256, 0, stream>>>(keys, inW + 1024, inB + 1024, nullptr, VB,
        2048, 512, 512, 512, 1536, 512, 0);
    mha_flash_k<<<1024, 256, 0, stream>>>(PA, KB, VB, PB);
    gemm_k<<<dim3(8, 128), 256, 0, stream>>>(PB, ca_out_w + (size_t)i * 512 * 512,
        ca_out_b + i * 512, cur1, cur1, 8192, 512, 512, 512, 512, 512, 0);
    { float* t = cur1; cur1 = cur2; cur2 = t; }
    // ---- FFN sublayer ----
    ln_kernel<<<8192, 256, 0, stream>>>(cur2, ln_ff_g + i * 512, ln_ff_b + i * 512, LN);
    gemm_k<<<dim3(32, 128), 256, 0, stream>>>(LN, ff_w1 + (size_t)i * 512 * 2048,
        ff_b1 + (size_t)i * 2048, nullptr, BIG, 8192, 2048, 512, 512, 2048, 2048, 1);
    gemm_k<<<dim3(8, 128), 256, 0, stream>>>(BIG, ff_w2 + (size_t)i * 2048 * 512,
        ff_b2 + i * 512, cur1, cur1, 8192, 512, 2048, 2048, 512, 512, 0);
    { float* t = cur1; cur1 = cur2; cur2 = t; }
  }
  add_k<<<(int)((SZ_X + 255) / 256), 256, 0, stream>>>(cur1, cur2, (float*)d_out, SZ_X);
}